// HSVTransform_62294205661351
// MI455X (gfx1250) — compile-verified
//
#include <hip/hip_runtime.h>

typedef __attribute__((ext_vector_type(4))) float v4f;

// (64, 3, 512, 512) fp32 RGB -> HSV. Pure streaming elementwise transform.
// Plane size = 512*512 = 262144 floats = 65536 float4s (power of two).
#define PLANE_F4    65536       // float4s per channel plane
#define PLANE_SHIFT 16          // log2(PLANE_F4)

__global__ __launch_bounds__(256) void
HSVTransform_62294205661351_kernel(const float* __restrict__ x,
                                   float* __restrict__ out,
                                   int n4_total)
{
    int j = blockIdx.x * blockDim.x + threadIdx.x;
    if (j >= n4_total) return;

    // j indexes float4 pixels across all 64 images: j = n * PLANE_F4 + p
    int n = j >> PLANE_SHIFT;           // image index
    int p = j & (PLANE_F4 - 1);         // float4 offset within a plane

    const v4f* __restrict__ xin = (const v4f*)x + (size_t)n * 3 * PLANE_F4 + p;
    v4f*       __restrict__ o   = (v4f*)out     + (size_t)n * 3 * PLANE_F4 + p;

    // 3 x global_load_b128 th:TH_LOAD_NT (403MB streamed once, > 192MB L2)
    v4f r = __builtin_nontemporal_load(xin);
    v4f g = __builtin_nontemporal_load(xin + PLANE_F4);
    v4f b = __builtin_nontemporal_load(xin + 2 * PLANE_F4);

    v4f h, s, v;

#pragma unroll
    for (int k = 0; k < 4; ++k) {
        float R = r[k], G = g[k], B = b[k];

        float mx = fmaxf(fmaxf(R, G), B);      // v_max3_num_f32
        float mn = fminf(fminf(R, G), B);      // v_min3_num_f32
        float d  = mx - mn;

        float ds = (d  == 0.0f) ? 1.0f : d;    // delta_safe
        float ms = (mx == 0.0f) ? 1.0f : mx;   // maxc_safe

        // v_rcp_f32 (~1 ulp) instead of IEEE div sequences: we are HBM-bound,
        // keep VALU/thread low so memory stays the only bottleneck.
        float invd = __builtin_amdgcn_rcpf(ds);
        float invm = __builtin_amdgcn_rcpf(ms);

        float S = (mx == 0.0f) ? 0.0f : d * invm;

        // hue: pick numerator and sector offset, then one FMA
        float num = (mx == R) ? (G - B) : (mx == G) ? (B - R) : (R - G);
        float off = (mx == R) ? 0.0f    : (mx == G) ? 2.0f    : 4.0f;
        float H   = fmaf(num, invd, off);

        // (h / 6) mod 1 with Python semantics (result in [0,1) even for h < 0)
        float t = H * (1.0f / 6.0f);
        t = t - floorf(t);
        H = (d == 0.0f) ? 0.0f : t;

        h[k] = H;
        s[k] = S;
        v[k] = mx;
    }

    // 3 x global_store_b128 th:TH_STORE_NT
    __builtin_nontemporal_store(h, o);
    __builtin_nontemporal_store(s, o + PLANE_F4);
    __builtin_nontemporal_store(v, o + 2 * PLANE_F4);
}

extern "C" void kernel_launch(void* const* d_in, const int* in_sizes, int n_in,
                              void* d_out, int out_size, void* d_ws, size_t ws_size,
                              hipStream_t stream)
{
    (void)in_sizes; (void)n_in; (void)out_size; (void)d_ws; (void)ws_size;

    const float* x   = (const float*)d_in[0];
    float*       out = (float*)d_out;

    const int n4_total = 64 * PLANE_F4;            // 4,194,304 threads
    const int block    = 256;                      // 8 wave32s
    const int grid     = (n4_total + block - 1) / block;

    HSVTransform_62294205661351_kernel<<<grid, block, 0, stream>>>(x, out, n4_total);
}